// GAT_5497558139164
// MI455X (gfx1250) — compile-verified
//
#include <hip/hip_runtime.h>
#include <math.h>

typedef __attribute__((ext_vector_type(2))) float v2f;
typedef __attribute__((ext_vector_type(8))) float v8f;

#define IN_F   128
#define H_F    32
#define OUT_F  16
#define HEADS  4
#define NEG_SLOPE 0.2f

static __device__ __forceinline__ float lrelu(float x) {
    return x > 0.0f ? x : NEG_SLOPE * x;
}

// Float atomic max via sign-split integer trick (any sign mix; -inf/0xFF.. init ok).
static __device__ __forceinline__ void atomicMaxF32(float* addr, float val) {
    if (!(val < 0.0f)) atomicMax((int*)addr, __float_as_int(val));
    else               atomicMin((unsigned int*)addr, (unsigned int)__float_as_int(val));
}

// src/dst for virtual edge list = [real edges | self loops]
static __device__ __forceinline__ void edge_nodes(const int* __restrict__ ei, int E,
                                                  int e, int& src, int& dst) {
    if (e < E) { src = ei[e]; dst = ei[E + e]; }
    else       { src = dst = e - E; }
}

__global__ void fill_f32(float* __restrict__ p, float v, int n) {
    int i = blockIdx.x * blockDim.x + threadIdx.x;
    if (i < n) p[i] = v;
}

// ---------------------------------------------------------------------------
// C[M,N] = A[M,K] @ B[K,N], row-major, V_WMMA_F32_16X16X4_F32.
// Block = 256 threads (8 waves). Block tile: 128 rows x NB cols.
// - B staged transposed in LDS: Bt[n][k], row pitch K+2 (keeps 8B alignment,
//   spreads banks). Lane's B fragment (k,k+1 at fixed col) = one ds_load_b64
//   with immediate offset.
// - A fragment (k,k+1 at fixed row) = one global_load_b64 with immediate
//   offset from a per-lane base pointer (K is compile-time).
// - Each wave computes a 16 x NB strip: per k-step 1 global b64 + (NB/16)
//   ds b64 + (NB/16) wmma.
// Fragment layouts per ISA 7.12.2 (f32, 16x16x4):
//   A: lanes 0-15 K={0,1}, lanes 16-31 K={2,3}, M=lane&15
//   B: mirrored, N=lane&15
//   C/D: VGPR r -> M = r + (lane<16?0:8), N = lane&15
// ---------------------------------------------------------------------------
template <int K, int NB>
__global__ __launch_bounds__(256) void gemm_wmma_f32(
    const float* __restrict__ A, const float* __restrict__ B, float* __restrict__ C,
    int M, int N) {
    constexpr int KP = K + 2;            // padded LDS row pitch (even -> 8B align)
    __shared__ float Bt[NB * KP];

    const int tid     = threadIdx.x;
    const int colBase = blockIdx.y * NB;

    // Stage B^T tile: coalesced global reads, strided LDS writes.
    for (int i = tid; i < K * NB; i += 256) {
        int k = i / NB, n = i % NB;
        Bt[n * KP + k] = B[(size_t)k * N + colBase + n];
    }
    __syncthreads();

    const int wave  = tid >> 5;
    const int lane  = tid & 31;
    const int row   = blockIdx.x * 128 + wave * 16 + (lane & 15);
    const int rowc  = row < M ? row : M - 1;        // clamp; no EXEC divergence
    const int khalf = (lane >> 4) << 1;             // 0 or 2

    const float* ap = A + (size_t)rowc * K + khalf;
    const float* bp = &Bt[(lane & 15) * KP + khalf];

    v8f acc[NB / 16] = {};
    #pragma unroll
    for (int kk = 0; kk < K; kk += 4) {
        v2f a = *(const v2f*)(ap + kk);             // global_load_b64, imm offset
        #pragma unroll
        for (int t = 0; t < NB / 16; ++t) {
            v2f b = *(const v2f*)(bp + t * 16 * KP + kk);   // ds_load_b64, imm offset
            acc[t] = __builtin_amdgcn_wmma_f32_16x16x4_f32(
                /*neg_a=*/false, a, /*neg_b=*/false, b,
                /*c_mod=*/(short)0, acc[t], /*reuse_a=*/false, /*reuse_b=*/false);
        }
    }

    const int mrow = blockIdx.x * 128 + wave * 16 + ((lane >> 4) << 3);  // +0 or +8
    const int col  = colBase + (lane & 15);
    #pragma unroll
    for (int t = 0; t < NB / 16; ++t) {
        #pragma unroll
        for (int r = 0; r < 8; ++r) {
            int m = mrow + r;
            if (m < M) C[(size_t)m * N + col + t * 16] = acc[t][r];
        }
    }
}

// al_src[n,h] = dot(h[n,h,:], a_src[h,:]) ; same for a_dst. One thread per (n,h).
template <int F>
__global__ void node_logits(const float* __restrict__ h,
                            const float* __restrict__ a_src,
                            const float* __restrict__ a_dst,
                            float* __restrict__ als, float* __restrict__ ald, int Nn) {
    int i = blockIdx.x * blockDim.x + threadIdx.x;
    if (i >= Nn * HEADS) return;
    int n  = i / HEADS;
    int hd = i % HEADS;
    const float* hp = h + (size_t)n * HEADS * F + hd * F;
    float s = 0.f, d = 0.f;
    #pragma unroll
    for (int f = 0; f < F; ++f) {
        float v = hp[f];
        s += v * a_src[hd * F + f];
        d += v * a_dst[hd * F + f];
    }
    als[i] = s;
    ald[i] = d;
}

// Pass 1: segment max of per-edge logits. One thread per (edge, head).
__global__ void edge_max_k(const int* __restrict__ ei,
                           const float* __restrict__ als, const float* __restrict__ ald,
                           float* __restrict__ m, int E, int Etot) {
    int i = blockIdx.x * blockDim.x + threadIdx.x;
    if (i >= Etot * HEADS) return;
    int e = i >> 2, hd = i & 3;
    int s, d;
    edge_nodes(ei, E, e, s, d);
    float v = lrelu(als[s * HEADS + hd] + ald[d * HEADS + hd]);
    atomicMaxF32(&m[d * HEADS + hd], v);
}

// Pass 2: segment sum of exp(e - max).
__global__ void edge_sum_k(const int* __restrict__ ei,
                           const float* __restrict__ als, const float* __restrict__ ald,
                           const float* __restrict__ m, float* __restrict__ denom,
                           int E, int Etot) {
    int i = blockIdx.x * blockDim.x + threadIdx.x;
    if (i >= Etot * HEADS) return;
    int e = i >> 2, hd = i & 3;
    int s, d;
    edge_nodes(ei, E, e, s, d);
    float v = lrelu(als[s * HEADS + hd] + ald[d * HEADS + hd]);
    atomicAdd(&denom[d * HEADS + hd], expf(v - m[d * HEADS + hd]));
}

// Pass 3: agg[dst,h,f] += alpha * hsrc[src,h,f]. One thread per (edge, feature);
// lanes cover consecutive f -> coalesced L2 gather + coalesced atomics.
template <int F>
__global__ void edge_agg_k(const int* __restrict__ ei,
                           const float* __restrict__ als, const float* __restrict__ ald,
                           const float* __restrict__ m, const float* __restrict__ denom,
                           const float* __restrict__ h, float* __restrict__ agg,
                           int E, int Etot) {
    int i = blockIdx.x * blockDim.x + threadIdx.x;
    if (i >= Etot * F) return;
    int e = i / F, f = i % F;
    int s, d;
    edge_nodes(ei, E, e, s, d);
    #pragma unroll
    for (int hd = 0; hd < HEADS; ++hd) {
        float v     = lrelu(als[s * HEADS + hd] + ald[d * HEADS + hd]);
        float alpha = expf(v - m[d * HEADS + hd]) / (denom[d * HEADS + hd] + 1e-16f);
        atomicAdd(&agg[(size_t)d * HEADS * F + hd * F + f],
                  alpha * h[(size_t)s * HEADS * F + hd * F + f]);
    }
}

// Mean over heads + bias (+optional ReLU).
template <int F, bool RELU>
__global__ void finalize_k(const float* __restrict__ agg, const float* __restrict__ bias,
                           float* __restrict__ out, int Nn) {
    int i = blockIdx.x * blockDim.x + threadIdx.x;
    if (i >= Nn * F) return;
    int n = i / F, f = i % F;
    float s = 0.f;
    #pragma unroll
    for (int hd = 0; hd < HEADS; ++hd) s += agg[(size_t)n * HEADS * F + hd * F + f];
    s = s * (1.0f / HEADS) + bias[f];
    if (RELU) s = fmaxf(s, 0.f);
    out[i] = s;
}

static inline int cdiv(long long a, long long b) { return (int)((a + b - 1) / b); }

extern "C" void kernel_launch(void* const* d_in, const int* in_sizes, int n_in,
                              void* d_out, int out_size, void* d_ws, size_t ws_size,
                              hipStream_t stream) {
    const float* x      = (const float*)d_in[0];
    const int*   ei     = (const int*)  d_in[1];
    const float* W1     = (const float*)d_in[2];
    const float* a_src1 = (const float*)d_in[3];
    const float* a_dst1 = (const float*)d_in[4];
    const float* b1     = (const float*)d_in[5];
    const float* W2     = (const float*)d_in[6];
    const float* a_src2 = (const float*)d_in[7];
    const float* a_dst2 = (const float*)d_in[8];
    const float* b2     = (const float*)d_in[9];
    float* out = (float*)d_out;

    const int Nn   = in_sizes[0] / IN_F;   // 50000
    const int E    = in_sizes[1] / 2;      // 800000
    const int Etot = E + Nn;               // + self loops

    // Workspace carve-up (floats). Total ~21.6M floats (~86 MB).
    float* ws    = (float*)d_ws;
    size_t o     = 0;
    float* h1    = ws + o; o += (size_t)Nn * HEADS * H_F;    // [N,4,32]
    float* agg1  = ws + o; o += (size_t)Nn * HEADS * H_F;
    float* hrel  = ws + o; o += (size_t)Nn * H_F;            // [N,32]
    float* h2    = ws + o; o += (size_t)Nn * HEADS * OUT_F;  // [N,4,16]
    float* agg2  = ws + o; o += (size_t)Nn * HEADS * OUT_F;
    float* als   = ws + o; o += (size_t)Nn * HEADS;
    float* ald   = ws + o; o += (size_t)Nn * HEADS;
    float* mbuf  = ws + o; o += (size_t)Nn * HEADS;
    float* dbuf  = ws + o; o += (size_t)Nn * HEADS;

    const int BLK = 256;
    const int nh  = Nn * HEADS;

    // ---------------- Layer 1 ----------------
    {
        dim3 grid(cdiv(Nn, 128), (HEADS * H_F) / 64);   // (391, 2)
        gemm_wmma_f32<IN_F, 64><<<grid, BLK, 0, stream>>>(x, W1, h1, Nn, HEADS * H_F);
    }
    node_logits<H_F><<<cdiv(nh, BLK), BLK, 0, stream>>>(h1, a_src1, a_dst1, als, ald, Nn);
    fill_f32<<<cdiv(nh, BLK), BLK, 0, stream>>>(mbuf, -INFINITY, nh);
    fill_f32<<<cdiv(nh, BLK), BLK, 0, stream>>>(dbuf, 0.0f, nh);
    fill_f32<<<cdiv((long long)Nn * HEADS * H_F, BLK), BLK, 0, stream>>>(
        agg1, 0.0f, Nn * HEADS * H_F);
    edge_max_k<<<cdiv((long long)Etot * HEADS, BLK), BLK, 0, stream>>>(ei, als, ald, mbuf, E, Etot);
    edge_sum_k<<<cdiv((long long)Etot * HEADS, BLK), BLK, 0, stream>>>(ei, als, ald, mbuf, dbuf, E, Etot);
    edge_agg_k<H_F><<<cdiv((long long)Etot * H_F, BLK), BLK, 0, stream>>>(
        ei, als, ald, mbuf, dbuf, h1, agg1, E, Etot);
    finalize_k<H_F, true><<<cdiv((long long)Nn * H_F, BLK), BLK, 0, stream>>>(agg1, b1, hrel, Nn);

    // ---------------- Layer 2 ----------------
    {
        dim3 grid(cdiv(Nn, 128), (HEADS * OUT_F) / 64); // (391, 1)
        gemm_wmma_f32<H_F, 64><<<grid, BLK, 0, stream>>>(hrel, W2, h2, Nn, HEADS * OUT_F);
    }
    node_logits<OUT_F><<<cdiv(nh, BLK), BLK, 0, stream>>>(h2, a_src2, a_dst2, als, ald, Nn);
    fill_f32<<<cdiv(nh, BLK), BLK, 0, stream>>>(mbuf, -INFINITY, nh);
    fill_f32<<<cdiv(nh, BLK), BLK, 0, stream>>>(dbuf, 0.0f, nh);
    fill_f32<<<cdiv((long long)Nn * HEADS * OUT_F, BLK), BLK, 0, stream>>>(
        agg2, 0.0f, Nn * HEADS * OUT_F);
    edge_max_k<<<cdiv((long long)Etot * HEADS, BLK), BLK, 0, stream>>>(ei, als, ald, mbuf, E, Etot);
    edge_sum_k<<<cdiv((long long)Etot * HEADS, BLK), BLK, 0, stream>>>(ei, als, ald, mbuf, dbuf, E, Etot);
    edge_agg_k<OUT_F><<<cdiv((long long)Etot * OUT_F, BLK), BLK, 0, stream>>>(
        ei, als, ald, mbuf, dbuf, h2, agg2, E, Etot);
    finalize_k<OUT_F, false><<<cdiv((long long)Nn * OUT_F, BLK), BLK, 0, stream>>>(agg2, b2, out, Nn);
}